// ReIdTotalLoss_12086037971168
// MI455X (gfx1250) — compile-verified
//
#include <hip/hip_runtime.h>
#include <math.h>

typedef __attribute__((ext_vector_type(2))) float v2f;
typedef __attribute__((ext_vector_type(8))) float v8f;

#define N_ 2048
#define D_ 1024
#define C_ 8192

__device__ __forceinline__ float relu_(float x) { return x > 0.f ? x : 0.f; }

// ---------------- workspace layout (floats) ----------------
// [0,2048)    : sq[i] = ||f_i||^2
// [2048,4096) : ap bits (uint, hardest-positive distance, monotone bitcast)
// [4096,6144) : an bits (uint, hardest-negative distance)
// [6144]      : acc_id
// [6145]      : acc_center

__global__ void k_init(float* ws) {
    int i = blockIdx.x * blockDim.x + threadIdx.x;
    unsigned* ap = (unsigned*)(ws + 2048);
    unsigned* an = (unsigned*)(ws + 4096);
    if (i < N_) { ap[i] = 0u; an[i] = 0x7F800000u; }   // 0.0f / +inf
    if (i == 0) { ws[6144] = 0.f; ws[6145] = 0.f; }
}

__global__ __launch_bounds__(256) void k_sq(const float* __restrict__ f,
                                            float* __restrict__ sq) {
    __shared__ float red[256];
    const int row = blockIdx.x, tid = threadIdx.x;
    const float4* r = (const float4*)(f + (size_t)row * D_);
    float4 v = r[tid];  // D_/4 == 256 == blockDim
    float s = v.x * v.x + v.y * v.y + v.z * v.z + v.w * v.w;
    red[tid] = s; __syncthreads();
    for (int st = 128; st > 0; st >>= 1) {
        if (tid < st) red[tid] += red[tid + st];
        __syncthreads();
    }
    if (!tid) sq[row] = red[0];
}

// Gram tile kernel: block = 8 waves, region 64 (i) x 128 (j).
// Wave w computes 4 stacked 16x16 tiles at columns jBase + w*16.
__global__ __launch_bounds__(256) void k_gram(const float* __restrict__ F,
                                              const float* __restrict__ sq,
                                              const int* __restrict__ labels,
                                              unsigned* __restrict__ ap,
                                              unsigned* __restrict__ an) {
    constexpr int KS = 36;                 // LDS row stride in floats (16B-aligned, bank-spread)
    __shared__ float As[64 * KS];
    __shared__ float Bs[128 * KS];

    const int tid  = threadIdx.x;
    const int wave = tid >> 5;
    const int lane = tid & 31;
    const int l15  = lane & 15;
    const int hi   = lane >> 4;            // K half-select for f32 wmma fragments
    const int iBase = blockIdx.y * 64;
    const int jBase = blockIdx.x * 128;

    v8f acc[4] = {};

    for (int k0 = 0; k0 < D_; k0 += 32) {
        __syncthreads();
        // stage 64x32 (As) and 128x32 (Bs) feature slabs as float4
        int idx = tid;
#pragma unroll
        for (int s = 0; s < 2; ++s, idx += 256) {
            int row = idx >> 3, c4 = (idx & 7) << 2;
            float4 v = *(const float4*)&F[(size_t)(iBase + row) * D_ + k0 + c4];
            *(float4*)&As[row * KS + c4] = v;
        }
        idx = tid;
#pragma unroll
        for (int s = 0; s < 4; ++s, idx += 256) {
            int row = idx >> 3, c4 = (idx & 7) << 2;
            float4 v = *(const float4*)&F[(size_t)(jBase + row) * D_ + k0 + c4];
            *(float4*)&Bs[row * KS + c4] = v;
        }
        __syncthreads();

#pragma unroll
        for (int kk = 0; kk < 32; kk += 4) {
            const int ko = kk + 2 * hi;    // lanes 0-15: K={kk,kk+1}; lanes 16-31: K={kk+2,kk+3}
            v2f b = *(const v2f*)&Bs[(wave * 16 + l15) * KS + ko];
#pragma unroll
            for (int m = 0; m < 4; ++m) {
                v2f a = *(const v2f*)&As[(m * 16 + l15) * KS + ko];
                acc[m] = __builtin_amdgcn_wmma_f32_16x16x4_f32(
                    false, a, false, b, (short)0, acc[m], false, false);
            }
        }
    }

    // Fused hard-mining epilogue: C/D layout => lane(l15)=N, VGPR v + 8*hi = M
    const int   j    = jBase + wave * 16 + l15;
    const float sqj  = sq[j];
    const int   labj = labels[j];
    const float inf_ = __uint_as_float(0x7F800000u);

#pragma unroll
    for (int m = 0; m < 4; ++m) {
#pragma unroll
        for (int v = 0; v < 8; ++v) {
            const int i = iBase + m * 16 + v + 8 * hi;   // uniform per half-wave
            const float g  = acc[m][v];
            const float d2 = sq[i] + sqj - 2.0f * g;
            const float d  = sqrtf(fmaxf(d2, 1e-12f));
            const bool same = (labels[i] == labj);
            float pos = same ? d : 0.0f;
            float neg = same ? inf_ : d;
#pragma unroll
            for (int o = 8; o; o >>= 1) {
                pos = fmaxf(pos, __shfl_xor(pos, o, 16));
                neg = fminf(neg, __shfl_xor(neg, o, 16));
            }
            if (l15 == 0) {
                atomicMax(&ap[i], __float_as_uint(pos));  // bit-order == float order (x>=0)
                atomicMin(&an[i], __float_as_uint(neg));
            }
        }
    }
}

__global__ __launch_bounds__(256) void k_ce(const float* __restrict__ logits,
                                            const int* __restrict__ labels,
                                            float* acc_id) {
    __shared__ float red[256];
    const int row = blockIdx.x, tid = threadIdx.x;
    const float* z = logits + (size_t)row * C_;

    float lmax = -3.402823466e38f, lsum = 0.f;
    for (int c = tid; c < C_; c += 256) { float v = z[c]; lmax = fmaxf(lmax, v); lsum += v; }

    red[tid] = lmax; __syncthreads();
    for (int s = 128; s > 0; s >>= 1) { if (tid < s) red[tid] = fmaxf(red[tid], red[tid + s]); __syncthreads(); }
    const float m = red[0]; __syncthreads();

    red[tid] = lsum; __syncthreads();
    for (int s = 128; s > 0; s >>= 1) { if (tid < s) red[tid] += red[tid + s]; __syncthreads(); }
    const float sumz = red[0]; __syncthreads();

    float lexp = 0.f;
    for (int c = tid; c < C_; c += 256) lexp += expf(z[c] - m);
    red[tid] = lexp; __syncthreads();
    for (int s = 128; s > 0; s >>= 1) { if (tid < s) red[tid] += red[tid + s]; __syncthreads(); }

    if (!tid) {
        const float lse = m + logf(red[0]);
        const float zy  = z[labels[row]];
        const float eps = 0.1f;
        const float contrib = -(1.f - eps) * (zy - lse)
                              - (eps / (float)C_) * (sumz - (float)C_ * lse);
        atomicAdd(acc_id, contrib / (float)N_);
    }
}

__global__ __launch_bounds__(256) void k_center(const float* __restrict__ f,
                                                const float* __restrict__ centers,
                                                const int* __restrict__ labels,
                                                float* acc_center) {
    __shared__ float red[256];
    const int row = blockIdx.x, tid = threadIdx.x;
    const float4* fr = (const float4*)(f + (size_t)row * D_);
    const float4* cr = (const float4*)(centers + (size_t)labels[row] * D_);
    float4 a = fr[tid], b = cr[tid];   // D_/4 == 256 == blockDim
    float dx = a.x - b.x, dy = a.y - b.y, dz = a.z - b.z, dw = a.w - b.w;
    red[tid] = dx * dx + dy * dy + dz * dz + dw * dw;
    __syncthreads();
    for (int s = 128; s > 0; s >>= 1) { if (tid < s) red[tid] += red[tid + s]; __syncthreads(); }
    if (!tid) {
        float v = fminf(fmaxf(red[0], 1e-12f), 1e12f);
        atomicAdd(acc_center, v / (float)N_);
    }
}

__global__ __launch_bounds__(256) void k_final(const float* ws, float* out) {
    __shared__ float red[256];
    const unsigned* ap = (const unsigned*)(ws + 2048);
    const unsigned* an = (const unsigned*)(ws + 4096);
    const int tid = threadIdx.x;
    float s = 0.f;
    for (int i = tid; i < N_; i += 256) {
        float dap = __uint_as_float(ap[i]);
        float dan = __uint_as_float(an[i]);
        s += relu_(dap - dan + 0.3f);
    }
    red[tid] = s; __syncthreads();
    for (int st = 128; st > 0; st >>= 1) { if (tid < st) red[tid] += red[tid + st]; __syncthreads(); }
    if (!tid)
        out[0] = red[0] / (float)N_      // W_METRIC * metric_loss
               + ws[6144]                // W_ID * id_loss
               + 5e-4f * ws[6145];       // W_CENTER * center_loss
}

extern "C" void kernel_launch(void* const* d_in, const int* in_sizes, int n_in,
                              void* d_out, int out_size, void* d_ws, size_t ws_size,
                              hipStream_t stream) {
    const float* features = (const float*)d_in[0];
    const float* logits   = (const float*)d_in[1];
    const float* centers  = (const float*)d_in[2];
    const int*   labels   = (const int*)d_in[3];
    float* ws  = (float*)d_ws;
    float* out = (float*)d_out;

    float*    sq = ws;
    unsigned* ap = (unsigned*)(ws + 2048);
    unsigned* an = (unsigned*)(ws + 4096);
    float* acc_id     = ws + 6144;
    float* acc_center = ws + 6145;

    k_init<<<(N_ + 255) / 256, 256, 0, stream>>>(ws);
    k_sq<<<N_, 256, 0, stream>>>(features, sq);
    k_gram<<<dim3(N_ / 128, N_ / 64), 256, 0, stream>>>(features, sq, labels, ap, an);
    k_ce<<<N_, 256, 0, stream>>>(logits, labels, acc_id);
    k_center<<<N_, 256, 0, stream>>>(features, centers, labels, acc_center);
    k_final<<<1, 256, 0, stream>>>(ws, out);
}